// L2Similarity_16664473109190
// MI455X (gfx1250) — compile-verified
//
#include <hip/hip_runtime.h>
#include <math.h>

// L2-similarity logits:  out[n][m] = -sqrt(max(||x_n||^2 + ||g_m||^2 - 2 x_n.g_m, 0))
// x: [8192,128] f32, g: [16384,128] f32, out: [8192,16384] f32 (512 MB, store-BW bound)
//
// MI455X reasoning: HBM floor = 512MB / 23.3TB/s ~ 22us; GEMM part is 34 GFLOP
// -> entirely output-bandwidth bound. Use fp32 WMMA (V_WMMA_F32_16X16X4_F32) to
// match reference precision (free at this arithmetic intensity), non-temporal
// stores for the write-once output, and register-level A reuse (1 A fragment x
// 4 B tiles per wave). Inputs (12 MB) live in the 192 MB L2, so redundant
// per-wave loads are cheap.

typedef __attribute__((ext_vector_type(2))) float v2f;
typedef __attribute__((ext_vector_type(8))) float v8f;

#define DIM    128
#define TN     4          // 16-wide N tiles per wave
#define WAVES  8          // waves per block -> 128 rows per block

__global__ __launch_bounds__(256)
void l2sim_wmma_kernel(const float* __restrict__ X,   // [Nrows, 128]
                       const float* __restrict__ G,   // [Mrows, 128]
                       float* __restrict__ Out,       // [Nrows, Mrows]
                       int Mout)                      // = Mrows (16384)
{
    const int lane = threadIdx.x & 31;
    const int wave = threadIdx.x >> 5;
    const int hi   = lane >> 4;          // 0: K%4 in {0,1}, 1: K%4 in {2,3}
    const int l16  = lane & 15;
    const int koff = hi * 2;

    const int m0 = blockIdx.y * (WAVES * 16) + wave * 16;  // output row base (x rows)
    const int n0 = blockIdx.x * (TN * 16);                 // output col base (g rows)

    // Identical per-lane addressing for A and B fragments (see analysis):
    const float* xrow = X + (size_t)(m0 + l16) * DIM + koff;
    const float* grow = G + (size_t)(n0 + l16) * DIM + koff;

    v8f acc[TN] = {};            // 4 accumulator tiles, 8 VGPRs each
    float xsq = 0.f;
    float gsq[TN] = {0.f, 0.f, 0.f, 0.f};

    #pragma unroll 8
    for (int k = 0; k < DIM; k += 4) {
        v2f a = *(const v2f*)(xrow + k);
        xsq = fmaf(a.x, a.x, fmaf(a.y, a.y, xsq));
        #pragma unroll
        for (int t = 0; t < TN; ++t) {
            v2f b = *(const v2f*)(grow + (size_t)t * 16 * DIM + k);
            gsq[t] = fmaf(b.x, b.x, fmaf(b.y, b.y, gsq[t]));
            // v_wmma_f32_16x16x4_f32: D = A(16x4) * B(4x16) + C
            acc[t] = __builtin_amdgcn_wmma_f32_16x16x4_f32(
                false, a, false, b, (short)0, acc[t], false, false);
        }
    }

    // Each lane holds half of its row's norm (k mod 4 halves); fold across lane^16.
    xsq += __shfl_xor(xsq, 16, 32);
    #pragma unroll
    for (int t = 0; t < TN; ++t) gsq[t] += __shfl_xor(gsq[t], 16, 32);

    // C layout (32-bit 16x16): VGPR j, lanes 0-15 -> row m0+j, lanes 16-31 -> row m0+j+8;
    // col = n0 + 16t + (lane&15).  gsq[t] is already lane-aligned with the column.
    #pragma unroll
    for (int t = 0; t < TN; ++t) {
        const int c = n0 + t * 16 + l16;
        #pragma unroll
        for (int j = 0; j < 8; ++j) {
            float xs = __shfl(xsq, j + 8 * hi, 32);  // broadcast row norm
            float d2 = fmaxf(xs + gsq[t] - 2.0f * acc[t][j], 0.0f);
            float v  = -sqrtf(d2);
            const int r = m0 + j + 8 * hi;
            // write-once 512MB stream: non-temporal to keep it out of L2
            __builtin_nontemporal_store(v, Out + (size_t)r * Mout + c);
        }
    }
}

extern "C" void kernel_launch(void* const* d_in, const int* in_sizes, int n_in,
                              void* d_out, int out_size, void* d_ws, size_t ws_size,
                              hipStream_t stream) {
    const float* X = (const float*)d_in[0];   // image_features [8192,128]
    const float* G = (const float*)d_in[1];   // gts            [16384,128]
    float* Out = (float*)d_out;               // [8192,16384]

    const int Nrows = in_sizes[0] / DIM;      // 8192
    const int Mrows = in_sizes[1] / DIM;      // 16384

    dim3 block(256, 1, 1);
    dim3 grid(Mrows / (TN * 16), Nrows / (WAVES * 16), 1);  // (256, 64)
    l2sim_wmma_kernel<<<grid, block, 0, stream>>>(X, G, Out, Mrows);
}